// GCN_61229053772176
// MI455X (gfx1250) — compile-verified
//
#include <hip/hip_runtime.h>
#include <hip/hip_bf16.h>
#include <math.h>

// ---------------------------------------------------------------------------
// GCN forward on gfx1250 (MI455X).
//   layer: y = dinv ⊙ (X @ W)          -> WMMA f32 16x16x4 GEMM (fp32 exact)
//          acc[dst] += y[src]          -> wave-per-edge, global f32 atomics (L2)
//          h = act(dinv ⊙ (acc + y) + b)  (self-loop folded in analytically)
//   pool:  per-graph mean via atomics, then tiny MLP + sigmoid head.
// Roofline: scatter traffic (~3.3 GB, L2-resident) dominates; GEMM is ~205 MB.
// ---------------------------------------------------------------------------

typedef __attribute__((ext_vector_type(2))) float v2f;
typedef __attribute__((ext_vector_type(8))) float v8f;

#define XS 130   // LDS row stride (floats) for the X stripe  (16 rows)
#define WS 136   // LDS row stride (floats) for W             (128 rows)

// ---------------- utility kernels ----------------

__global__ void k_fill(float* __restrict__ p, long long n, float v) {
    long long i = (long long)blockIdx.x * blockDim.x + threadIdx.x;
    if (i < n) p[i] = v;
}

__global__ void k_deg(const long long* __restrict__ dst, float* __restrict__ deg,
                      long long E) {
    long long e = (long long)blockIdx.x * blockDim.x + threadIdx.x;
    if (e < E) atomicAdd(&deg[dst[e]], 1.0f);
}

__global__ void k_rsqrt(float* __restrict__ dinv, long long N) {
    long long i = (long long)blockIdx.x * blockDim.x + threadIdx.x;
    if (i < N) dinv[i] = rsqrtf(dinv[i]);   // deg >= 1 always (self-loop)
}

__global__ void k_count(const long long* __restrict__ batch, float* __restrict__ cnt,
                        long long N) {
    long long i = (long long)blockIdx.x * blockDim.x + threadIdx.x;
    if (i < N) atomicAdd(&cnt[batch[i]], 1.0f);
}

// ---------------- WMMA GEMM:  Y[N,128] = dinv ⊙ (X[N,128] @ W[128,128]) ------

__global__ __launch_bounds__(256)
void k_gemm_scale(const float* __restrict__ X, const float* __restrict__ Wm,
                  const float* __restrict__ dinv, float* __restrict__ Y, int N) {
    __shared__ float sX[16 * XS];    //  ~8.3 KB
    __shared__ float sW[128 * WS];   // ~69.6 KB
    const int tid  = threadIdx.x;
    const int row0 = blockIdx.x * 16;

    // Stage W (row-major, padded stride) — shared by all 8 waves.
    for (int i = tid; i < 128 * 128; i += 256) {
        int r = i >> 7, c = i & 127;
        sW[r * WS + c] = Wm[i];
    }
    // Stage the 16-row X stripe (zero-pad past N).
    for (int i = tid; i < 16 * 128; i += 256) {
        int r = i >> 7, c = i & 127;
        int row = row0 + r;
        sX[r * XS + c] = (row < N) ? X[(long long)row * 128 + c] : 0.0f;
    }
    __syncthreads();

    const int wave = tid >> 5;        // 8 waves -> 8 column tiles of 16
    const int lane = tid & 31;
    const int m    = lane & 15;       // A row / BCD column index
    const int half = lane >> 4;       // K-pair select (A/B), M+8 select (C/D)
    const int col  = wave * 16 + m;

    v8f acc = {0.f,0.f,0.f,0.f,0.f,0.f,0.f,0.f};
    #pragma unroll
    for (int kk = 0; kk < 32; ++kk) {          // K = 128 in steps of 4
        const int k = kk * 4 + half * 2;
        // A 16x4: lane m holds K = {k, k+1}
        v2f a = { sX[m * XS + k], sX[m * XS + k + 1] };
        // B 4x16: lane n=col holds rows K = {k, k+1}
        v2f b = { sW[k * WS + col], sW[(k + 1) * WS + col] };
        acc = __builtin_amdgcn_wmma_f32_16x16x4_f32(
            /*neg_a=*/false, a, /*neg_b=*/false, b,
            /*c_mod=*/(short)0, acc, /*reuse_a=*/false, /*reuse_b=*/false);
    }

    // C/D layout: VGPR r -> M = r + 8*half, N = lane&15.  Scale rows by dinv.
    #pragma unroll
    for (int r = 0; r < 8; ++r) {
        int row = row0 + half * 8 + r;
        if (row < N)
            Y[(long long)row * 128 + col] = acc[r] * dinv[row];
    }
}

// ---------------- edge scatter: acc[dst] += y[src] (wave per edge) -----------

__global__ __launch_bounds__(256)
void k_scatter(const float* __restrict__ Y, float* __restrict__ acc,
               const long long* __restrict__ ei, long long E) {
    long long e = (long long)blockIdx.x * 8 + (threadIdx.x >> 5);
    if (e >= E) return;
    const long long s = ei[e];        // src
    const long long d = ei[E + e];    // dst
    const int lane = threadIdx.x & 31;
    const float4 v = ((const float4*)(Y + s * 128))[lane];   // 32*16B = 512B row
    float* p = acc + d * 128 + lane * 4;
    atomicAdd(p + 0, v.x);
    atomicAdd(p + 1, v.y);
    atomicAdd(p + 2, v.z);
    atomicAdd(p + 3, v.w);
}

// ---------------- epilogues ----------------

// h = relu(dinv ⊙ (acc + y) + b), written in place over acc.
__global__ void k_epilogue_relu(float* __restrict__ acc, const float* __restrict__ y,
                                const float* __restrict__ dinv,
                                const float* __restrict__ bias, long long total) {
    long long idx = (long long)blockIdx.x * blockDim.x + threadIdx.x;
    if (idx >= total) return;
    long long i = idx >> 7;
    int c = (int)(idx & 127);
    float v = dinv[i] * (acc[idx] + y[idx]) + bias[c];
    acc[idx] = fmaxf(v, 0.0f);
}

// layer-2 output goes straight into the per-graph pooled sums.
__global__ void k_epilogue_pool(const float* __restrict__ acc, const float* __restrict__ y,
                                const float* __restrict__ dinv,
                                const float* __restrict__ bias,
                                const long long* __restrict__ batch,
                                float* __restrict__ pool, long long total) {
    long long idx = (long long)blockIdx.x * blockDim.x + threadIdx.x;
    if (idx >= total) return;
    long long i = idx >> 7;
    int c = (int)(idx & 127);
    float v = dinv[i] * (acc[idx] + y[idx]) + bias[c];
    atomicAdd(&pool[(int)batch[i] * 128 + c], v);
}

// ---------------- MLP head: 64 graphs, 128 -> 64 -> 1, sigmoid ----------------

__global__ __launch_bounds__(64)
void k_mlp(const float* __restrict__ pool, const float* __restrict__ cnt,
           const float* __restrict__ Wl1, const float* __restrict__ bl1,
           const float* __restrict__ Wl2, const float* __restrict__ bl2,
           float* __restrict__ out) {
    const int g = threadIdx.x;                 // 64 graphs
    const float invc = 1.0f / fmaxf(cnt[g], 1.0f);
    float p[128];
    #pragma unroll 8
    for (int k = 0; k < 128; ++k) p[k] = pool[g * 128 + k] * invc;
    float z = 0.0f;
    for (int c = 0; c < 64; ++c) {
        float h = bl1[c];
        #pragma unroll 8
        for (int k = 0; k < 128; ++k) h += p[k] * Wl1[k * 64 + c];
        z += fmaxf(h, 0.0f) * Wl2[c];
    }
    z += bl2[0];
    out[g] = 1.0f / (1.0f + expf(-z));
}

// ---------------- launcher ----------------

extern "C" void kernel_launch(void* const* d_in, const int* in_sizes, int n_in,
                              void* d_out, int out_size, void* d_ws, size_t ws_size,
                              hipStream_t stream) {
    const float*      x     = (const float*)d_in[0];
    const long long*  ei    = (const long long*)d_in[1];   // [2, E] int64
    const long long*  batch = (const long long*)d_in[2];   // [N]    int64
    const float*      W1    = (const float*)d_in[3];
    const float*      b1    = (const float*)d_in[4];
    const float*      W2    = (const float*)d_in[5];
    const float*      b2    = (const float*)d_in[6];
    const float*      Wl1   = (const float*)d_in[7];
    const float*      bl1   = (const float*)d_in[8];
    const float*      Wl2   = (const float*)d_in[9];
    const float*      bl2   = (const float*)d_in[10];
    float*            out   = (float*)d_out;

    const long long N = (long long)in_sizes[0] / 128;
    const long long E = (long long)in_sizes[1] / 2;
    const long long tot = N * 128;

    // workspace layout (256B-aligned regions)
    char* ws = (char*)d_ws;
    size_t o = 0;
    auto al = [](size_t v) { return (v + 255) & ~(size_t)255; };
    float* dinv = (float*)(ws + o); o = al(o + (size_t)N * 4);
    float* bufA = (float*)(ws + o); o = al(o + (size_t)tot * 4);   // y (scaled GEMM out)
    float* bufB = (float*)(ws + o); o = al(o + (size_t)tot * 4);   // acc / h
    float* pool = (float*)(ws + o); o = al(o + (64 * 128 + 64) * 4);
    float* cnt  = pool + 64 * 128;
    (void)ws_size; (void)n_in; (void)out_size;

    #define CDIV(a, b) ((unsigned)(((a) + (b) - 1) / (b)))
    const dim3 blk(256);

    // degrees (self-loop => start at 1), then dinv = rsqrt(deg)
    k_fill <<<CDIV(N, 256),   blk, 0, stream>>>(dinv, N, 1.0f);
    k_deg  <<<CDIV(E, 256),   blk, 0, stream>>>(ei + E, dinv, E);
    k_rsqrt<<<CDIV(N, 256),   blk, 0, stream>>>(dinv, N);

    // ---- layer 1 ----
    k_fill      <<<CDIV(tot, 256), blk, 0, stream>>>(bufB, tot, 0.0f);
    k_gemm_scale<<<CDIV(N, 16),    blk, 0, stream>>>(x, W1, dinv, bufA, (int)N);
    k_scatter   <<<CDIV(E, 8),     blk, 0, stream>>>(bufA, bufB, ei, E);
    k_epilogue_relu<<<CDIV(tot, 256), blk, 0, stream>>>(bufB, bufA, dinv, b1, tot);

    // ---- layer 2 (bufB currently holds h1) ----
    k_gemm_scale<<<CDIV(N, 16),    blk, 0, stream>>>(bufB, W2, dinv, bufA, (int)N);
    k_fill      <<<CDIV(tot, 256), blk, 0, stream>>>(bufB, tot, 0.0f);
    k_fill      <<<CDIV(64 * 128 + 64, 256), blk, 0, stream>>>(pool, 64 * 128 + 64, 0.0f);
    k_count     <<<CDIV(N, 256),   blk, 0, stream>>>(batch, cnt, N);
    k_scatter   <<<CDIV(E, 8),     blk, 0, stream>>>(bufA, bufB, ei, E);
    k_epilogue_pool<<<CDIV(tot, 256), blk, 0, stream>>>(bufB, bufA, dinv, b2, batch, pool, tot);

    // ---- head ----
    k_mlp<<<1, 64, 0, stream>>>(pool, cnt, Wl1, bl1, Wl2, bl2, out);
    #undef CDIV
}